// Layer_60035052863925
// MI455X (gfx1250) — compile-verified
//
#include <hip/hip_runtime.h>
#include <math.h>

// ---------------- problem constants (from reference) ----------------
#define NFK   4      // FACTOR_K
#define DIN   64     // DIM_IN
#define DK    16     // DIM_K
#define NU    100000
#define NI    50000
#define NT    10000

typedef __attribute__((ext_vector_type(2))) float v2f;
typedef __attribute__((ext_vector_type(8))) float v8f;

__device__ __forceinline__ float leaky_f(float x) { return x >= 0.f ? x : 0.2f * x; }

__device__ __forceinline__ void atomic_add_f32(float* p, float v) {
    __hip_atomic_fetch_add(p, v, __ATOMIC_RELAXED, __HIP_MEMORY_SCOPE_AGENT);
}

// D = A(16x4) * B(4x16) + C, fp32 WMMA (CDNA5)
__device__ __forceinline__ v8f wmma16x16x4(v2f a, v2f b, v8f c) {
    return __builtin_amdgcn_wmma_f32_16x16x4_f32(false, a, false, b, (short)0, c,
                                                 false, false);
}

// ---------------------------------------------------------------
// ego[t] = l2n(leaky(emb @ Wc)), Wc[d][k*16+f] = Wtk[t][k][d][f]
// One wave computes a 16-node x 16-col (one factor k) tile.
// ---------------------------------------------------------------
__global__ void fac_kernel(const float* __restrict__ emb,
                           const float* __restrict__ Wtk_t, // [4][64][16]
                           float* __restrict__ ego, int N) {
    int gid  = blockIdx.x * blockDim.x + threadIdx.x;
    int wave = gid >> 5;
    int lane = gid & 31;
    int total = (N / 16) * 4;
    if (wave >= total) return;
    int rowTile = wave >> 2;
    int k       = wave & 3;          // factor == column tile
    int n0 = rowTile * 16;
    int M  = lane & 15;
    int hi = lane >> 4;              // 0: lanes 0-15, 1: lanes 16-31
    const float* Wk = Wtk_t + (size_t)k * (DIN * DK); // [d][f]

    v8f c = {};
    #pragma unroll
    for (int db = 0; db < DIN; db += 4) {
        int d0 = db + hi * 2;
        v2f a, b;
        a.x = emb[(size_t)(n0 + M) * DIN + d0];
        a.y = emb[(size_t)(n0 + M) * DIN + d0 + 1];
        b.x = Wk[(size_t)d0 * DK + M];
        b.y = Wk[(size_t)(d0 + 1) * DK + M];
        c = wmma16x16x4(a, b, c);
    }
    // leaky + per-factor L2 norm (norm over f == across 16 lanes of each half)
    #pragma unroll
    for (int i = 0; i < 8; i++) {
        float x = leaky_f(c[i]);
        float s = x * x;
        s += __shfl_xor(s, 1, 32);
        s += __shfl_xor(s, 2, 32);
        s += __shfl_xor(s, 4, 32);
        s += __shfl_xor(s, 8, 32);
        float nrm = fmaxf(sqrtf(s), 1e-12f);
        int row = n0 + i + hi * 8;
        ego[(size_t)row * 64 + k * 16 + M] = x / nrm;
    }
}

// ---------------------------------------------------------------
// Phase A: per edge attention logit -> ex = exp(vals); den[row] += ex
// ---------------------------------------------------------------
__global__ void edge_vals_kernel(const int* __restrict__ rows,
                                 const int* __restrict__ cols, int E,
                                 const float* __restrict__ src,   // new[srcType]
                                 const float* __restrict__ dst,   // ego[dstType]
                                 const float* __restrict__ at_e,  // [4][32]
                                 const float* __restrict__ r_e,   // [4][nrow]
                                 int nrow,
                                 float* __restrict__ ex_buf,
                                 float* __restrict__ den) {
    int e = blockIdx.x * blockDim.x + threadIdx.x;
    if (e >= E) return;
    int row = rows[e], col = cols[e];
    const float4* sp = (const float4*)(src + (size_t)row * 64);
    const float4* dp = (const float4*)(dst + (size_t)col * 64);
    float vals = 0.f;
    #pragma unroll
    for (int k = 0; k < NFK; k++) {
        const float* au = at_e + k * 32;
        const float* ai = au + 16;
        float ets = 0.f;
        #pragma unroll
        for (int q = 0; q < 4; q++) {
            float4 s4 = sp[k * 4 + q];
            float4 d4 = dp[k * 4 + q];
            ets += s4.x * au[q * 4 + 0] + s4.y * au[q * 4 + 1] +
                   s4.z * au[q * 4 + 2] + s4.w * au[q * 4 + 3];
            ets += d4.x * ai[q * 4 + 0] + d4.y * ai[q * 4 + 1] +
                   d4.z * ai[q * 4 + 2] + d4.w * ai[q * 4 + 3];
        }
        float rk = r_e[(size_t)k * nrow + row];
        vals += fmaxf(ets, 0.f) * rk;
    }
    float exv = expf(vals);   // softmax shift-invariant; vals bounded, safe
    ex_buf[e] = exv;
    atomic_add_f32(&den[row], exv);
}

// ---------------------------------------------------------------
// Phase B: z[row] += attn * ego_dst[col]   (4 threads per edge)
// ---------------------------------------------------------------
__global__ void edge_scatter_kernel(const int* __restrict__ rows,
                                    const int* __restrict__ cols, int E,
                                    const float* __restrict__ dst,
                                    const float* __restrict__ ex_buf,
                                    const float* __restrict__ den,
                                    float* __restrict__ z) {
    int idx = blockIdx.x * blockDim.x + threadIdx.x;
    int e = idx >> 2;
    if (e >= E) return;
    int part = idx & 3;
    int row = rows[e], col = cols[e];
    float attn = ex_buf[e] / den[row];
    const float* dp = dst + (size_t)col * 64 + part * 16;
    float* zp = z + (size_t)row * 64 + part * 16;
    #pragma unroll
    for (int j = 0; j < 16; j++) atomic_add_f32(&zp[j], dp[j] * attn);
}

// ---------------------------------------------------------------
// Phase C1: z <- leaky(z) @ W  in place, via WMMA.
// Rows m = row*4 + k  (z[m*16+f] == z[row*64 + k*16 + f]).
// ---------------------------------------------------------------
__global__ void zW_kernel(float* __restrict__ z, const float* __restrict__ W,
                          int Mrows) {
    int gid  = blockIdx.x * blockDim.x + threadIdx.x;
    int wave = gid >> 5;
    int lane = gid & 31;
    if (wave >= Mrows / 16) return;
    int m0 = wave * 16;
    int M  = lane & 15;
    int hi = lane >> 4;
    v8f c = {};
    #pragma unroll
    for (int kb = 0; kb < 16; kb += 4) {
        int kk = kb + hi * 2;
        v2f a, b;
        a.x = leaky_f(z[(size_t)(m0 + M) * 16 + kk]);
        a.y = leaky_f(z[(size_t)(m0 + M) * 16 + kk + 1]);
        b.x = W[kk * 16 + M];
        b.y = W[(kk + 1) * 16 + M];
        c = wmma16x16x4(a, b, c);
    }
    #pragma unroll
    for (int i = 0; i < 8; i++) {
        int m = m0 + i + hi * 8;
        z[(size_t)m * 16 + M] = c[i];
    }
}

// ---------------------------------------------------------------
// Phase C2: r[k,row] = softmax_k(tanh(z2)·q); acc[row] += z2 * r
// ---------------------------------------------------------------
__global__ void r_acc_kernel(const float* __restrict__ z2,
                             const float* __restrict__ q, int nrow,
                             float* __restrict__ r_e,
                             float* __restrict__ acc) {
    int row = blockIdx.x * blockDim.x + threadIdx.x;
    if (row >= nrow) return;
    const float* zp = z2 + (size_t)row * 64;
    float zv[64], s[4];
    #pragma unroll
    for (int k = 0; k < NFK; k++) {
        float sk = 0.f;
        #pragma unroll
        for (int f = 0; f < 16; f++) {
            float v = zp[k * 16 + f];
            zv[k * 16 + f] = v;
            sk += tanhf(v) * q[f];
        }
        s[k] = sk;
    }
    float m = fmaxf(fmaxf(s[0], s[1]), fmaxf(s[2], s[3]));
    float e[4], sum = 0.f;
    #pragma unroll
    for (int k = 0; k < NFK; k++) { e[k] = expf(s[k] - m); sum += e[k]; }
    float* ap = acc + (size_t)row * 64;
    #pragma unroll
    for (int k = 0; k < NFK; k++) {
        float rk = e[k] / sum;
        r_e[(size_t)k * nrow + row] = rk;
        #pragma unroll
        for (int f = 0; f < 16; f++) ap[k * 16 + f] += zv[k * 16 + f] * rk;
    }
}

// ---------------------------------------------------------------
// new = l2n(ego + acc)   (one thread per (node, k))
// ---------------------------------------------------------------
__global__ void newfac_kernel(const float* __restrict__ ego,
                              const float* __restrict__ acc,
                              float* __restrict__ out, int N) {
    int idx = blockIdx.x * blockDim.x + threadIdx.x;
    if (idx >= N * NFK) return;
    size_t base = (size_t)idx * 16;
    float x[16], s = 0.f;
    #pragma unroll
    for (int f = 0; f < 16; f++) {
        float v = ego[base + f] + acc[base + f];
        x[f] = v; s += v * v;
    }
    float nrm = fmaxf(sqrtf(s), 1e-12f);
    #pragma unroll
    for (int f = 0; f < 16; f++) out[base + f] = x[f] / nrm;
}

__global__ void fill_kernel(float* __restrict__ p, float v, int n) {
    int i = blockIdx.x * blockDim.x + threadIdx.x;
    if (i < n) p[i] = v;
}

// ---------------------------------------------------------------
extern "C" void kernel_launch(void* const* d_in, const int* in_sizes, int n_in,
                              void* d_out, int out_size, void* d_ws, size_t ws_size,
                              hipStream_t stream) {
    const float* u_emb = (const float*)d_in[0];
    const float* i_emb = (const float*)d_in[1];
    const float* t_emb = (const float*)d_in[2];
    const float* Wtk   = (const float*)d_in[3];   // [3][4][64][16]
    const float* at    = (const float*)d_in[4];   // [6][4][32]
    const float* W     = (const float*)d_in[5];   // [16][16]
    const float* qr    = (const float*)d_in[6];   // [6][16]

    const int* edges[6]; int E[6];
    for (int e = 0; e < 6; e++) {
        edges[e] = (const int*)d_in[7 + e];
        E[e] = in_sizes[7 + e] / 2;
    }

    const int NN[3]    = {NU, NI, NT};
    const int NROWS[6] = {NU, NI, NU, NT, NI, NT};
    const int SRC[6]   = {0, 1, 0, 2, 1, 2};   // INDEX[e][0]
    const int DST[6]   = {1, 0, 2, 0, 2, 1};   // INDEX[e][1]
    const float* embs[3] = {u_emb, i_emb, t_emb};

    // ---- workspace carve (floats) ----
    float* ws = (float*)d_ws;
    size_t off = 0;
    float* ego[3]; for (int t = 0; t < 3; t++) { ego[t] = ws + off; off += (size_t)NN[t] * 64; }
    float* acc[3]; for (int t = 0; t < 3; t++) { acc[t] = ws + off; off += (size_t)NN[t] * 64; }
    float* zbuf = ws + off; off += (size_t)NU * 64;
    float* rbuf[6]; size_t r_total = 0;
    for (int e = 0; e < 6; e++) { rbuf[e] = ws + off; off += (size_t)4 * NROWS[e]; r_total += (size_t)4 * NROWS[e]; }
    float* exbuf = ws + off; off += 500000;
    float* den   = ws + off; off += NU;
    (void)ws_size; (void)out_size; (void)n_in;

    // "new" lives directly in d_out (layout matches output exactly)
    float* newp[3];
    newp[0] = (float*)d_out;
    newp[1] = newp[0] + (size_t)NU * 64;
    newp[2] = newp[1] + (size_t)NI * 64;

    // 1) ego = l2n(leaky(emb @ Wtk)); new = ego
    for (int t = 0; t < 3; t++) {
        int waves = (NN[t] / 16) * 4;
        int thr   = waves * 32;
        fac_kernel<<<(thr + 255) / 256, 256, 0, stream>>>(
            embs[t], Wtk + (size_t)t * 4 * 64 * 16, ego[t], NN[t]);
        hipMemcpyAsync(newp[t], ego[t], (size_t)NN[t] * 64 * sizeof(float),
                       hipMemcpyDeviceToDevice, stream);
    }

    // 2) r = 1/4 (rbuf is contiguous)
    {
        int n = (int)r_total;
        fill_kernel<<<(n + 255) / 256, 256, 0, stream>>>(rbuf[0], 0.25f, n);
    }

    // 3) two message-passing iterations
    for (int iter = 0; iter < 2; iter++) {
        for (int t = 0; t < 3; t++)
            hipMemsetAsync(acc[t], 0, (size_t)NN[t] * 64 * sizeof(float), stream);

        for (int e = 0; e < 6; e++) {
            int nrow = NROWS[e];
            const float* src = newp[SRC[e]];
            const float* dst = ego[DST[e]];
            const int* rows = edges[e];
            const int* cols = edges[e] + E[e];

            hipMemsetAsync(den, 0, (size_t)nrow * sizeof(float), stream);
            hipMemsetAsync(zbuf, 0, (size_t)nrow * 64 * sizeof(float), stream);

            edge_vals_kernel<<<(E[e] + 255) / 256, 256, 0, stream>>>(
                rows, cols, E[e], src, dst, at + (size_t)e * 128, rbuf[e], nrow,
                exbuf, den);

            edge_scatter_kernel<<<(E[e] * 4 + 255) / 256, 256, 0, stream>>>(
                rows, cols, E[e], dst, exbuf, den, zbuf);

            int Mrows = nrow * 4;
            int waves = Mrows / 16;
            zW_kernel<<<(waves * 32 + 255) / 256, 256, 0, stream>>>(zbuf, W, Mrows);

            r_acc_kernel<<<(nrow + 255) / 256, 256, 0, stream>>>(
                zbuf, qr + (size_t)e * 16, nrow, rbuf[e], acc[SRC[e]]);
        }

        for (int t = 0; t < 3; t++) {
            int n = NN[t] * 4;
            newfac_kernel<<<(n + 255) / 256, 256, 0, stream>>>(
                ego[t], acc[t], newp[t], NN[t]);
        }
    }
}